// GCNRegression_8727373545871
// MI455X (gfx1250) — compile-verified
//
#include <hip/hip_runtime.h>

typedef __attribute__((ext_vector_type(2))) float v2f;
typedef __attribute__((ext_vector_type(4))) float v4f;
typedef __attribute__((ext_vector_type(8))) float v8f;

#define N_NODES 100000
#define N_EDGES 1600000
#define N_FEAT  128
#define HIDDEN  64

// ---------------------------------------------------------------------------
// Async global->LDS staging (CDNA5 GLOBAL_LOAD_ASYNC_TO_LDS_B128, ASYNCcnt).
// Builtin signature (from hipcc diagnostics): param0 is
//   'int __vector(4) __device__ *'  == v4i addrspace(1)*  (AS1 = __device__),
// so we cast through integers to the exact pointer types. Generic->AS via
// integer is valid on gfx1250: LDS aperture maps addr[31:0] directly to the
// LDS offset; global generic == global.
// ---------------------------------------------------------------------------
#if defined(__has_builtin)
#if __has_builtin(__builtin_amdgcn_global_load_async_to_lds_b128) && \
    __has_builtin(__builtin_amdgcn_s_wait_asynccnt)
#define USE_ASYNC_G2L 1
#endif
#endif
#ifndef USE_ASYNC_G2L
#define USE_ASYNC_G2L 0
#endif

typedef int v4i_vs __attribute__((vector_size(16)));
typedef __attribute__((address_space(1))) v4i_vs* as1_v4i_p;
typedef __attribute__((address_space(3))) v4i_vs* as3_v4i_p;

__device__ static inline void g2l_b128(const float* g, float* l) {
#if USE_ASYNC_G2L
    __builtin_amdgcn_global_load_async_to_lds_b128(
        (as1_v4i_p)(uintptr_t)g,
        (as3_v4i_p)(unsigned int)(uintptr_t)l,
        0, 0);
#else
    *(v4f*)l = *(const v4f*)g;
#endif
}

__device__ static inline void g2l_wait() {
#if USE_ASYNC_G2L
    __builtin_amdgcn_s_wait_asynccnt(0);
#endif
}

// ---------------------------------------------------------------------------
// Small utility kernels
// ---------------------------------------------------------------------------
__global__ __launch_bounds__(256) void set_val_kernel(float* __restrict__ p, int n, float v) {
    int i = blockIdx.x * blockDim.x + threadIdx.x;
    if (i < n) p[i] = v;
}

__global__ __launch_bounds__(256) void zero4_kernel(float* __restrict__ p, int n4) {
    int i = blockIdx.x * blockDim.x + threadIdx.x;
    if (i < n4) ((v4f*)p)[i] = (v4f){0.f, 0.f, 0.f, 0.f};
}

// deg[dst] += ew  (deg pre-initialized to 1.0 for the self loop)
__global__ __launch_bounds__(256) void deg_accum_kernel(const int* __restrict__ dst,
                                                        const float* __restrict__ ew,
                                                        float* __restrict__ deg) {
    int e = blockIdx.x * blockDim.x + threadIdx.x;
    if (e < N_EDGES) unsafeAtomicAdd(&deg[dst[e]], ew[e]);
}

// dis = deg > 0 ? rsqrt(deg) : 0   (in place)
__global__ __launch_bounds__(256) void finalize_dis_kernel(float* __restrict__ dis) {
    int i = blockIdx.x * blockDim.x + threadIdx.x;
    if (i < N_NODES) {
        float d = dis[i];
        dis[i] = (d > 0.f) ? rsqrtf(d) : 0.f;
    }
}

// norm[e] = dis[src] * ew * dis[dst]   (shared by both GCN layers)
__global__ __launch_bounds__(256) void edge_norm_kernel(const int* __restrict__ src,
                                                        const int* __restrict__ dst,
                                                        const float* __restrict__ ew,
                                                        const float* __restrict__ dis,
                                                        float* __restrict__ norm) {
    int e = blockIdx.x * blockDim.x + threadIdx.x;
    if (e < N_EDGES) norm[e] = dis[src[e]] * ew[e] * dis[dst[e]];
}

// ---------------------------------------------------------------------------
// WMMA fp32 GEMM:  Y[M x 64] = X[M x K] @ W[64 x K]^T      (K = 128 or 64)
//  * W staged once per block into LDS (row stride K+4: bank-conflict-free,
//    16B aligned) via async global->LDS when available.
//  * One wave per 16-row tile of Y, 4 v8f accumulators covering N=0..63.
//  * A operand software-pipelined one k-step ahead; K templated so the loop
//    fully unrolls and loads hoist across the WMMAs.
// A layout (16x4 f32): lane<16 -> K=k,k+1 ; lane>=16 -> K=k+2,k+3 (float2)
// B layout (4x16 f32): lane<16 -> rows k,k+1 ; lane>=16 -> k+2,k+3; col=lane&15
// C/D layout: VGPR j -> row rowbase + j + 8*(lane>=16), col = lane&15
// ---------------------------------------------------------------------------
template <int K>
__global__ __launch_bounds__(256) void gemm_xwt_wmma_kernel(const float* __restrict__ X,
                                                            const float* __restrict__ W,
                                                            float* __restrict__ Y,
                                                            int M) {
    constexpr int KP = K + 4;                 // padded LDS row stride (floats)
    __shared__ float sW[HIDDEN * KP];

    // ---- stage W[64][K] -> sW[64][KP] (16B chunks, coalesced) ----
    for (int idx = threadIdx.x; idx < HIDDEN * (K >> 2); idx += 256) {
        int row = idx / (K >> 2);
        int c   = (idx - row * (K >> 2)) << 2;
        g2l_b128(W + (size_t)row * K + c, &sW[row * KP + c]);
    }
    g2l_wait();
    __syncthreads();

    const int wave = threadIdx.x >> 5;
    const int lane = threadIdx.x & 31;
    const int ln   = lane & 15;
    const int hi   = lane >> 4;                      // 0 or 1
    const int rowbase = (blockIdx.x * 8 + wave) * 16;
    if (rowbase >= M) return;                        // wave-uniform (after barrier)

    v8f acc0 = {}, acc1 = {}, acc2 = {}, acc3 = {};

    const float* ap = X + (size_t)(rowbase + ln) * K + 2 * hi;
    const float* sb = sW + ln * KP + 2 * hi;

    v2f a = *(const v2f*)ap;                         // prefetch k = 0
#pragma unroll
    for (int k = 0; k < K; k += 4) {
        v2f acur = a;
        const int kn = (k + 4 < K) ? (k + 4) : k;    // clamp, resolved at unroll
        a = *(const v2f*)(ap + kn);                  // prefetch next k-step
        v2f b0 = *(const v2f*)(sb + 0 * 16 * KP + k);
        v2f b1 = *(const v2f*)(sb + 1 * 16 * KP + k);
        v2f b2 = *(const v2f*)(sb + 2 * 16 * KP + k);
        v2f b3 = *(const v2f*)(sb + 3 * 16 * KP + k);
        acc0 = __builtin_amdgcn_wmma_f32_16x16x4_f32(false, acur, false, b0, (short)0, acc0, false, false);
        acc1 = __builtin_amdgcn_wmma_f32_16x16x4_f32(false, acur, false, b1, (short)0, acc1, false, false);
        acc2 = __builtin_amdgcn_wmma_f32_16x16x4_f32(false, acur, false, b2, (short)0, acc2, false, false);
        acc3 = __builtin_amdgcn_wmma_f32_16x16x4_f32(false, acur, false, b3, (short)0, acc3, false, false);
    }

    float* yp = Y + (size_t)(rowbase + 8 * hi) * HIDDEN + ln;
#pragma unroll
    for (int j = 0; j < 8; ++j) {
        yp[(size_t)j * HIDDEN +  0] = acc0[j];
        yp[(size_t)j * HIDDEN + 16] = acc1[j];
        yp[(size_t)j * HIDDEN + 32] = acc2[j];
        yp[(size_t)j * HIDDEN + 48] = acc3[j];
    }
}

// ---------------------------------------------------------------------------
// Edge scatter-add: agg[dst,:] += norm[e] * xw[src,:]
// 16 threads per edge, float4 per thread (HIDDEN = 64). Whole feature buffer
// (25.6 MB) + norms (6.4 MB) fit in the 192 MB L2, so default temporal policy.
// ---------------------------------------------------------------------------
__global__ __launch_bounds__(256) void edge_scatter_kernel(const int* __restrict__ src,
                                                           const int* __restrict__ dst,
                                                           const float* __restrict__ norm,
                                                           const float* __restrict__ xw,
                                                           float* __restrict__ agg) {
    unsigned t = blockIdx.x * blockDim.x + threadIdx.x;   // N_EDGES * 16 threads
    int e = (int)(t >> 4);
    int c = ((int)t & 15) << 2;
    if (e >= N_EDGES) return;
    int s = src[e], d = dst[e];
    float nm = norm[e];
    v4f v = *(const v4f*)(xw + (size_t)s * HIDDEN + c);
    float* ag = agg + (size_t)d * HIDDEN + c;
    unsafeAtomicAdd(ag + 0, nm * v.x);
    unsafeAtomicAdd(ag + 1, nm * v.y);
    unsafeAtomicAdd(ag + 2, nm * v.z);
    unsafeAtomicAdd(ag + 3, nm * v.w);
}

// out[i,:] = relu(agg[i,:] + dis[i]^2 * xw[i,:] + bias[:])
__global__ __launch_bounds__(256) void self_relu_kernel(const float* __restrict__ agg,
                                                        const float* __restrict__ xw,
                                                        const float* __restrict__ dis,
                                                        const float* __restrict__ bias,
                                                        float* __restrict__ out) {
    unsigned t = blockIdx.x * blockDim.x + threadIdx.x;   // N_NODES * 16 threads
    int i = (int)(t >> 4);
    int c = ((int)t & 15) << 2;
    if (i >= N_NODES) return;
    float d2 = dis[i] * dis[i];
    v4f a = *(const v4f*)(agg  + (size_t)i * HIDDEN + c);
    v4f x = *(const v4f*)(xw   + (size_t)i * HIDDEN + c);
    v4f b = *(const v4f*)(bias + c);
    v4f r;
    r.x = fmaxf(fmaf(d2, x.x, a.x) + b.x, 0.f);
    r.y = fmaxf(fmaf(d2, x.y, a.y) + b.y, 0.f);
    r.z = fmaxf(fmaf(d2, x.z, a.z) + b.z, 0.f);
    r.w = fmaxf(fmaf(d2, x.w, a.w) + b.w, 0.f);
    *(v4f*)(out + (size_t)i * HIDDEN + c) = r;
}

// out[i] = dot(h[i,:], W3[0,:]) + b3
__global__ __launch_bounds__(256) void proj_out_kernel(const float* __restrict__ h,
                                                       const float* __restrict__ W3,
                                                       const float* __restrict__ b3,
                                                       float* __restrict__ out) {
    int i = blockIdx.x * blockDim.x + threadIdx.x;
    if (i >= N_NODES) return;
    const v4f* hp = (const v4f*)(h + (size_t)i * HIDDEN);
    const v4f* wp = (const v4f*)W3;
    float s = 0.f;
#pragma unroll
    for (int q = 0; q < 16; ++q) {
        v4f a = hp[q], b = wp[q];
        s += a.x * b.x + a.y * b.y + a.z * b.z + a.w * b.w;
    }
    out[i] = s + b3[0];
}

// ---------------------------------------------------------------------------
// Host launch sequence
// ---------------------------------------------------------------------------
static inline size_t align512(size_t x) { return (x + 511) & ~(size_t)511; }

extern "C" void kernel_launch(void* const* d_in, const int* in_sizes, int n_in,
                              void* d_out, int out_size, void* d_ws, size_t ws_size,
                              hipStream_t stream) {
    const float* x   = (const float*)d_in[0];
    const int*   ei  = (const int*)  d_in[1];   // [2, N_EDGES] flattened
    const float* ew  = (const float*)d_in[2];
    /* d_in[3] = batch (unused: single graph) */
    const float* W1  = (const float*)d_in[4];
    const float* b1  = (const float*)d_in[5];
    const float* W2  = (const float*)d_in[6];
    const float* b2  = (const float*)d_in[7];
    const float* W3  = (const float*)d_in[8];
    const float* b3  = (const float*)d_in[9];
    float*       out = (float*)d_out;

    const int* src = ei;
    const int* dst = ei + N_EDGES;

    // Workspace carve-up (~58 MB total)
    char*  ws   = (char*)d_ws;
    size_t off  = 0;
    float* dis  = (float*)(ws + off); off = align512(off + (size_t)N_NODES * 4);
    float* norm = (float*)(ws + off); off = align512(off + (size_t)N_EDGES * 4);
    float* B1   = (float*)(ws + off); off = align512(off + (size_t)N_NODES * HIDDEN * 4);
    float* B2   = (float*)(ws + off);

    const int nodeBlk   = (N_NODES + 255) / 256;                 // 391
    const int edgeBlk   = (N_EDGES + 255) / 256;                 // 6250
    const int scatBlk   = (N_EDGES * 16) / 256;                  // 100000
    const int featBlk   = (N_NODES * 16 + 255) / 256;            // 6250
    const int zeroBlk   = (N_NODES * HIDDEN / 4 + 255) / 256;    // 6250
    const int gemmBlk   = ((N_NODES / 16) + 7) / 8;              // 782 blocks x 8 waves

    // Degree / normalization (shared across both layers)
    set_val_kernel<<<nodeBlk, 256, 0, stream>>>(dis, N_NODES, 1.0f);
    deg_accum_kernel<<<edgeBlk, 256, 0, stream>>>(dst, ew, dis);
    finalize_dis_kernel<<<nodeBlk, 256, 0, stream>>>(dis);
    edge_norm_kernel<<<edgeBlk, 256, 0, stream>>>(src, dst, ew, dis, norm);

    // ---- Layer 1: xw1 = x @ W1^T -> B1; agg -> B2; h1 = relu(...) -> B1
    gemm_xwt_wmma_kernel<N_FEAT><<<gemmBlk, 256, 0, stream>>>(x, W1, B1, N_NODES);
    zero4_kernel<<<zeroBlk, 256, 0, stream>>>(B2, N_NODES * HIDDEN / 4);
    edge_scatter_kernel<<<scatBlk, 256, 0, stream>>>(src, dst, norm, B1, B2);
    self_relu_kernel<<<featBlk, 256, 0, stream>>>(B2, B1, dis, b1, B1);

    // ---- Layer 2: xw2 = h1 @ W2^T -> B2; agg -> B1; h2 = relu(...) -> B1
    gemm_xwt_wmma_kernel<HIDDEN><<<gemmBlk, 256, 0, stream>>>(B1, W2, B2, N_NODES);
    zero4_kernel<<<zeroBlk, 256, 0, stream>>>(B1, N_NODES * HIDDEN / 4);
    edge_scatter_kernel<<<scatBlk, 256, 0, stream>>>(src, dst, norm, B2, B1);
    self_relu_kernel<<<featBlk, 256, 0, stream>>>(B1, B2, dis, b2, B1);

    // ---- Output projection
    proj_out_kernel<<<nodeBlk, 256, 0, stream>>>(B1, W3, b3, out);
}